// PTSModel_60696477827781
// MI455X (gfx1250) — compile-verified
//
#include <hip/hip_runtime.h>
#include <stdint.h>

#define TOPK      100
#define VMAX      50257
#define NTHREADS  1024
#define HBINS     2048          // top-11-bit coarse histogram
#define CAP       8192          // candidate buffer capacity

typedef __attribute__((ext_vector_type(4))) unsigned int u32x4;
typedef __attribute__((ext_vector_type(8))) int          i32x8;
typedef __attribute__((ext_vector_type(4))) int          i32x4;

// monotone map: larger float <-> larger u32  (3 VALU ops, branch-free)
__device__ __forceinline__ unsigned ordmap(float f) {
    unsigned u = __float_as_uint(f);
    unsigned m = (unsigned)((int)u >> 31) | 0x80000000u;
    return u ^ m;
}
__device__ __forceinline__ float unordmap(unsigned u) {
    unsigned s = (u & 0x80000000u) ? (u ^ 0x80000000u) : ~u;
    return __uint_as_float(s);
}

__global__ __launch_bounds__(NTHREADS, 1)
void pts_topk_softmax_kernel(const float* __restrict__ inp,
                             const long long* __restrict__ tokens,
                             const float* __restrict__ W1, const float* __restrict__ b1,
                             const float* __restrict__ W2, const float* __restrict__ b2,
                             const float* __restrict__ W3, const float* __restrict__ b3,
                             float* __restrict__ out, int V)
{
    // Full row resident in WGP LDS (CDNA5: up to 320 KB per workgroup). ~243 KB total.
    __shared__ __align__(16) float s_row[VMAX];     // 201 KB
    __shared__ unsigned s_hist[HBINS];              // 8 KB
    __shared__ unsigned s_gsum[64];
    __shared__ __align__(16) unsigned s_cand[CAP];  // 32 KB
    __shared__ unsigned s_sorted[TOPK];
    __shared__ float    s_h1[8], s_h2[8];
    __shared__ float    s_part[NTHREADS / 32];
    __shared__ unsigned s_ctl[4];                   // [0]=b*, [2]=append cnt
    __shared__ float    s_scal[2];                  // [0]=ca, [1]=cb

    const int tid = threadIdx.x;
    const int row = blockIdx.x;
    const float* rowp = inp + (size_t)row * (size_t)V;

    // chunk size: ~V/4, multiple of 4 so float4 LDS reads stay aligned
    const int cs = ((V >> 2) + 3) & ~3;

    // histogram a sub-range [lo,hi) of the LDS row (float4-vectorized)
    auto histRange = [&](int lo, int hi) {
        if (lo >= hi) return;
        const int n4 = (hi - lo) >> 2;
        const float4* b4 = reinterpret_cast<const float4*>(s_row + lo);
        for (int i = tid; i < n4; i += NTHREADS) {
            float4 v = b4[i];
            atomicAdd(&s_hist[ordmap(v.x) >> 21], 1u);
            atomicAdd(&s_hist[ordmap(v.y) >> 21], 1u);
            atomicAdd(&s_hist[ordmap(v.z) >> 21], 1u);
            atomicAdd(&s_hist[ordmap(v.w) >> 21], 1u);
        }
        for (int i = lo + (n4 << 2) + tid; i < hi; i += NTHREADS)
            atomicAdd(&s_hist[ordmap(s_row[i]) >> 21], 1u);
    };

    // ------- 1) issue 4 chunked TDM DMAs (row -> LDS); overlap with histogram -------
#if __has_builtin(__builtin_amdgcn_tensor_load_to_lds)
    if (tid < 32) {  // wave 0 issues TDM ops; they complete in order per wave
        const unsigned base_lds = (unsigned)(uintptr_t)(void*)&s_row[0];
        const i32x4 gz4 = {0, 0, 0, 0};
        const i32x8 gz8 = {0, 0, 0, 0, 0, 0, 0, 0};
        for (int c = 0; c < 4; ++c) {
            int lo = c * cs;
            int hi = min(V, lo + cs);
            unsigned un = (unsigned)(hi > lo ? hi - lo : 0);
            unsigned long long ga = (unsigned long long)(uintptr_t)(rowp + lo);

            u32x4 g0;
            g0.x = 1u;                                    // count=1 (valid user D#)
            g0.y = base_lds + (unsigned)lo * 4u;          // lds_addr
            g0.z = (unsigned)(ga & 0xFFFFFFFFu);          // global_addr[31:0]
            g0.w = (unsigned)((ga >> 32) & 0x01FFFFFFu)   // global_addr[56:32]
                 | (2u << 30);                            // type = 2

            i32x8 g1;
            g1[0] = (int)(2u << 16);                      // data_size = 4 bytes
            g1[1] = (int)((un & 0xFFFFu) << 16);          // tensor_dim0[15:0]
            g1[2] = (int)(((un >> 16) & 0xFFFFu) | (1u << 16)); // dim0 hi | tensor_dim1=1
            g1[3] = (int)((un & 0xFFFFu) << 16);          // tile_dim0 = n
            g1[4] = 1;                                    // tile_dim1=1, tile_dim2=0
            g1[5] = (int)un;                              // tensor_dim0_stride
            g1[6] = 0;
            g1[7] = 0;
            __builtin_amdgcn_tensor_load_to_lds(g0, g1, gz4, gz4, gz8, 0);
        }
    }
#else
    for (int i = tid; i < V; i += NTHREADS) s_row[i] = rowp[i];
#endif

    // zero histogram / counters while the DMA streams
    for (int i = tid; i < HBINS; i += NTHREADS) s_hist[i] = 0u;
    if (tid == 0) s_ctl[2] = 0u;

    // ------- 2) coarse histogram, chunk-by-chunk as the TDM completes -------
#if __has_builtin(__builtin_amdgcn_tensor_load_to_lds)
    if (tid < 32) __builtin_amdgcn_s_wait_tensorcnt(3);
    __syncthreads();
    histRange(0, min(V, cs));
    if (tid < 32) __builtin_amdgcn_s_wait_tensorcnt(2);
    __syncthreads();
    histRange(cs, min(V, 2 * cs));
    if (tid < 32) __builtin_amdgcn_s_wait_tensorcnt(1);
    __syncthreads();
    histRange(2 * cs, min(V, 3 * cs));
    if (tid < 32) __builtin_amdgcn_s_wait_tensorcnt(0);
    __syncthreads();
    histRange(3 * cs, V);
#else
    __syncthreads();
    histRange(0, V);
#endif
    __syncthreads();

    // ------- 3) find bin b* containing the 100th-largest (parallel 2-level scan) ---
    if (tid < 64) {
        unsigned s = 0;
        for (int b = 0; b < 32; ++b) s += s_hist[tid * 32 + b];
        s_gsum[tid] = s;
    }
    __syncthreads();
    if (tid == 0) {
        unsigned acc = 0; int G = 0;
        for (int g = 63; g >= 0; --g) {
            unsigned c = s_gsum[g];
            if (acc + c >= TOPK) { G = g; break; }
            acc += c;
        }
        int bstar = G * 32;
        for (int b = G * 32 + 31; b >= G * 32; --b) {
            unsigned c = s_hist[b];
            if (acc + c >= TOPK) { bstar = b; break; }
            acc += c;
        }
        s_ctl[0] = (unsigned)bstar;
    }
    __syncthreads();
    const unsigned thr = s_ctl[0] << 21;   // candidates: ord >= thr  (superset of top-100)

    // ------- 4) compact candidates (one vectorized full pass) -------
    {
        const int n4 = V >> 2;
        const float4* b4 = reinterpret_cast<const float4*>(s_row);
        for (int i = tid; i < n4; i += NTHREADS) {
            float4 v = b4[i];
            unsigned u0 = ordmap(v.x), u1 = ordmap(v.y), u2 = ordmap(v.z), u3 = ordmap(v.w);
            if (u0 >= thr) { unsigned p = atomicAdd(&s_ctl[2], 1u); if (p < CAP) s_cand[p] = u0; }
            if (u1 >= thr) { unsigned p = atomicAdd(&s_ctl[2], 1u); if (p < CAP) s_cand[p] = u1; }
            if (u2 >= thr) { unsigned p = atomicAdd(&s_ctl[2], 1u); if (p < CAP) s_cand[p] = u2; }
            if (u3 >= thr) { unsigned p = atomicAdd(&s_ctl[2], 1u); if (p < CAP) s_cand[p] = u3; }
        }
        for (int i = (n4 << 2) + tid; i < V; i += NTHREADS) {
            unsigned u = ordmap(s_row[i]);
            if (u >= thr) { unsigned p = atomicAdd(&s_ctl[2], 1u); if (p < CAP) s_cand[p] = u; }
        }
    }
    __syncthreads();
    const int cnt = (int)min(s_ctl[2], (unsigned)CAP);

    // ------- 5) exact rank-sort of the small candidate set (descending) -------
    for (int c = tid; c < cnt; c += NTHREADS) {
        unsigned my = s_cand[c];
        int rank = 0;
        for (int j = 0; j < cnt; ++j) {
            unsigned o = s_cand[j];
            rank += (int)((o > my) || (o == my && j < c));
        }
        if (rank < TOPK) s_sorted[rank] = my;
    }
    __syncthreads();

    // ------- 6) tiny MLP -> temperature -------
    if (tid < 5) {
        float acc = b1[tid];
        for (int k = 0; k < TOPK; ++k) acc += unordmap(s_sorted[k]) * W1[tid * TOPK + k];
        s_h1[tid] = fmaxf(acc, 0.0f);
    }
    __syncthreads();
    if (tid < 5) {
        float acc = b2[tid];
        for (int k = 0; k < 5; ++k) acc += s_h1[k] * W2[tid * 5 + k];
        s_h2[tid] = fmaxf(acc, 0.0f);
    }
    __syncthreads();
    if (tid == 0) {
        float acc = b3[0];
        for (int k = 0; k < 5; ++k) acc += s_h2[k] * W3[k];
        float x    = fabsf(acc);
        float sp   = (x > 20.0f) ? x : log1pf(expf(x));   // softplus
        float temp = fmaxf(sp, 1e-5f);
        float m    = unordmap(s_sorted[0]);               // row max = top-1
        float it   = 1.0f / temp;
        const float L2E = 1.44269504088896f;
        s_scal[0] = it * L2E;                             // ca
        s_scal[1] = -m * it * L2E;                        // cb: exp((x-m)*it)=exp2(x*ca+cb)
    }
    __syncthreads();

    // ------- 7) softmax sum (fma+exp2 per element) + token gather -------
    const float ca = s_scal[0], cb = s_scal[1];
    float part = 0.0f;
    {
        const int n4 = V >> 2;
        const float4* b4 = reinterpret_cast<const float4*>(s_row);
        for (int i = tid; i < n4; i += NTHREADS) {
            float4 v = b4[i];
            part += exp2f(fmaf(v.x, ca, cb));
            part += exp2f(fmaf(v.y, ca, cb));
            part += exp2f(fmaf(v.z, ca, cb));
            part += exp2f(fmaf(v.w, ca, cb));
        }
        for (int i = (n4 << 2) + tid; i < V; i += NTHREADS)
            part += exp2f(fmaf(s_row[i], ca, cb));
    }
    for (int off = 16; off > 0; off >>= 1)                // wave32 reduction
        part += __shfl_xor(part, off, 32);
    if ((tid & 31) == 0) s_part[tid >> 5] = part;
    __syncthreads();
    if (tid == 0) {
        float sum = 0.0f;
        for (int w = 0; w < NTHREADS / 32; ++w) sum += s_part[w];
        long long tok = tokens[row];
        float xt = s_row[(int)tok];
        out[row] = exp2f(fmaf(xt, ca, cb)) / sum;
    }
}

extern "C" void kernel_launch(void* const* d_in, const int* in_sizes, int n_in,
                              void* d_out, int out_size, void* d_ws, size_t ws_size,
                              hipStream_t stream) {
    const float*     inp    = (const float*)d_in[0];
    const long long* tokens = (const long long*)d_in[1];
    const float*     W1     = (const float*)d_in[2];
    const float*     b1     = (const float*)d_in[3];
    const float*     W2     = (const float*)d_in[4];
    const float*     b2     = (const float*)d_in[5];
    const float*     W3     = (const float*)d_in[6];
    const float*     b3     = (const float*)d_in[7];
    float*           out    = (float*)d_out;

    const int N = in_sizes[1];              // tokens: one per row
    const int V = in_sizes[0] / N;          // vocab size (50257)

    pts_topk_softmax_kernel<<<N, NTHREADS, 0, stream>>>(
        inp, tokens, W1, b1, W2, b2, W3, b3, out, V);
}